// LightGCN_handle_50044958933433
// MI455X (gfx1250) — compile-verified
//
#include <hip/hip_runtime.h>

#define NUM_USERS  100000
#define NUM_ITEMS  50000
#define N_NODES    150000
#define DIM        64
#define N_LAYERS   3
#define N_EDGES    3000000
#define BATCH_U    2048

typedef __attribute__((ext_vector_type(2))) float v2f;
typedef __attribute__((ext_vector_type(8))) float v8f;

// ---------------- init: x = acc = concat(user_emb, item_emb), float4 vectorized
__global__ __launch_bounds__(256) void lg_init(const float4* __restrict__ ue,
                                               const float4* __restrict__ ie,
                                               float4* __restrict__ x,
                                               float4* __restrict__ acc,
                                               int n4u, int n4tot) {
    int i = blockIdx.x * 256 + threadIdx.x;
    if (i >= n4tot) return;
    float4 v = (i < n4u) ? ue[i] : ie[i - n4u];
    x[i] = v;
    acc[i] = v;
}

// ---------------- zero a float4 buffer
__global__ __launch_bounds__(256) void lg_zero(float4* __restrict__ p, int n4) {
    int i = blockIdx.x * 256 + threadIdx.x;
    if (i < n4) p[i] = make_float4(0.f, 0.f, 0.f, 0.f);
}

// ---------------- SpMM scatter: one wave per edge, lane handles 2 dims.
// Gather of x_in[col] row (256B) is one fully-coalesced wave read; scatter via
// global_atomic_add_f32 lands in L2 (node tables fit in 192MB L2).
__global__ __launch_bounds__(256) void lg_spmm(const float* __restrict__ vals,
                                               const int*   __restrict__ rows,
                                               const int*   __restrict__ cols,
                                               const float* __restrict__ xin,
                                               float*       __restrict__ xout,
                                               int nE) {
    int e = blockIdx.x * 8 + (threadIdx.x >> 5);
    if (e >= nE) return;
    int lane = threadIdx.x & 31;
    float v = vals[e];
    long c = (long)cols[e] * DIM;
    long r = (long)rows[e] * DIM;
    float2 m = *(const float2*)(xin + c + lane * 2);
    atomicAdd(xout + r + lane * 2 + 0, v * m.x);
    atomicAdd(xout + r + lane * 2 + 1, v * m.y);
}

// ---------------- acc += x (float4)
__global__ __launch_bounds__(256) void lg_accadd(float4* __restrict__ acc,
                                                 const float4* __restrict__ x, int n4) {
    int i = blockIdx.x * 256 + threadIdx.x;
    if (i >= n4) return;
    float4 a = acc[i]; float4 b = x[i];
    a.x += b.x; a.y += b.y; a.z += b.z; a.w += b.w;
    acc[i] = a;
}

// ---------------- gather batch users, fold full 1/16 scale (1/4 per operand)
__global__ __launch_bounds__(256) void lg_gather(const float* __restrict__ acc,
                                                 const int*   __restrict__ users,
                                                 float*       __restrict__ ug, int total) {
    int i = blockIdx.x * 256 + threadIdx.x;
    if (i >= total) return;
    int b = i >> 6, d = i & 63;
    ug[i] = acc[(long)users[b] * DIM + d] * 0.0625f;
}

// ---------------- rating GEMM via V_WMMA_F32_16X16X4_F32
// One wave per 16x16 output tile; K=64 -> 16 chained K=4 WMMAs.
// A fragment (16x4 f32): lane<16 -> K={kb,kb+1}, lane>=16 -> K={kb+2,kb+3}, M=lane%16.
// B fragment (4x16 f32): symmetric, N=lane%16; B[k][n] = items[n][k] -> contiguous float2.
// C/D: VGPR r holds M = r + 8*(lane>=16), N = lane%16.
__global__ __launch_bounds__(256) void lg_gemm(const float* __restrict__ U,   // 2048 x 64 (pre-scaled 1/16)
                                               const float* __restrict__ I,   // 50000 x 64
                                               float*       __restrict__ out) // 2048 x 50000
{
    const int NT = NUM_ITEMS / 16;                 // 3125 N-tiles
    int wave = threadIdx.x >> 5;
    long waveId = (long)blockIdx.x * 8 + wave;
    int tn = (int)(waveId % NT);
    int tm = (int)(waveId / NT);

    int lane = threadIdx.x & 31;
    int half = lane >> 4;                          // 0 or 1
    int l16  = lane & 15;
    int m = tm * 16 + l16;
    int n = tn * 16 + l16;
    int koff = half * 2;

    const float* urow = U + (long)m * DIM;
    const float* irow = I + (long)n * DIM;

    v8f c = {};
#pragma unroll
    for (int kb = 0; kb < DIM; kb += 4) {
        v2f a = *(const v2f*)(urow + kb + koff);
        v2f b = *(const v2f*)(irow + kb + koff);
        c = __builtin_amdgcn_wmma_f32_16x16x4_f32(false, a, false, b,
                                                  (short)0, c, false, false);
    }
#pragma unroll
    for (int r = 0; r < 8; ++r) {
        int mm = tm * 16 + r + half * 8;
        out[(long)mm * NUM_ITEMS + n] = c[r];
    }
}

extern "C" void kernel_launch(void* const* d_in, const int* in_sizes, int n_in,
                              void* d_out, int out_size, void* d_ws, size_t ws_size,
                              hipStream_t stream) {
    const float* user_emb  = (const float*)d_in[0];
    const float* item_emb  = (const float*)d_in[1];
    const float* edge_vals = (const float*)d_in[2];
    const int*   edge_rows = (const int*)d_in[3];
    const int*   edge_cols = (const int*)d_in[4];
    const int*   users     = (const int*)d_in[5];
    float* out = (float*)d_out;

    const long NFLOAT = (long)N_NODES * DIM;       // 9,600,000 floats per node buffer
    float* ws = (float*)d_ws;
    float* xa  = ws;                               // x ping
    float* xb  = ws + NFLOAT;                      // x pong
    float* acc = ws + 2 * NFLOAT;                  // accumulator
    float* ug  = ws + 3 * NFLOAT;                  // gathered users, 2048*64

    const int n4u   = NUM_USERS * DIM / 4;         // 1,600,000
    const int n4tot = (int)(NFLOAT / 4);           // 2,400,000
    const int eblocks = N_EDGES / 8;               // 375,000

    // init: xa = acc = concat(user_emb, item_emb)
    lg_init<<<(n4tot + 255) / 256, 256, 0, stream>>>(
        (const float4*)user_emb, (const float4*)item_emb,
        (float4*)xa, (float4*)acc, n4u, n4tot);

    float* cur = xa;
    float* nxt = xb;
    for (int layer = 0; layer < N_LAYERS; ++layer) {
        lg_zero<<<(n4tot + 255) / 256, 256, 0, stream>>>((float4*)nxt, n4tot);
        lg_spmm<<<eblocks, 256, 0, stream>>>(edge_vals, edge_rows, edge_cols,
                                             cur, nxt, N_EDGES);
        lg_accadd<<<(n4tot + 255) / 256, 256, 0, stream>>>((float4*)acc,
                                                           (const float4*)nxt, n4tot);
        float* t = cur; cur = nxt; nxt = t;
    }

    // gather batch users with folded 1/16 scale
    const int gtot = BATCH_U * DIM;                // 131,072
    lg_gather<<<(gtot + 255) / 256, 256, 0, stream>>>(acc, users, ug, gtot);

    // rating = ug @ items^T ; items = acc[NUM_USERS:]
    const float* items = acc + (long)NUM_USERS * DIM;
    const long tiles = (long)(BATCH_U / 16) * (NUM_ITEMS / 16);  // 400,000 waves
    const int gblocks = (int)(tiles / 8);                        // 50,000 blocks, exact
    lg_gemm<<<gblocks, 256, 0, stream>>>(ug, items, out);
}